// param_distance_7980049236292
// MI455X (gfx1250) — compile-verified
//
#include <hip/hip_runtime.h>

typedef float v2f __attribute__((ext_vector_type(2)));
typedef float v8f __attribute__((ext_vector_type(8)));

#define KCAND 8
#define DFEAT 64

// One wave32 handles a tile of 16 points. The D=64 L1-distance reduction is
// done with V_WMMA_F32_16X16X4_F32 (A = |t - a| 16x4 slab, B = ones column),
// accumulating exact f32 sums in the C matrix, column 0.
// Candidate data is double-buffered in registers so 16 loads stay in flight
// while the WMMA chain consumes the previous candidate's buffer.
__global__ __launch_bounds__(256) void l1_argmin_gather_kernel(
    const float* __restrict__ tensor,
    const float* __restrict__ agg,
    float* __restrict__ out,
    int P)  // number of points = B*N
{
  const int lane = threadIdx.x & 31;
  const int waveInBlock = threadIdx.x >> 5;
  const int tile = blockIdx.x * 8 + waveInBlock;  // 16 points per tile
  if (tile * 16 >= P) return;

  const int m  = lane & 15;   // point-within-tile (also N column role)
  const int hi = lane >> 4;   // 0: features 4i+{0,1}; 1: features 4i+{2,3}
  const size_t point = (size_t)tile * 16 + m;
  const size_t kstride = (size_t)P * DFEAT;

  // Preload this lane's 32 tensor features (float2 per WMMA step).
  const float* tptr = tensor + point * DFEAT + hi * 2;
  v2f t[16];
#pragma unroll
  for (int i = 0; i < 16; ++i)
    t[i] = *(const v2f*)(tptr + i * 4);

  // B matrix: ones in column 0 of the 4x16 operand -> row-sum into D[:,0].
  v2f ones;
  ones.x = (m == 0) ? 1.0f : 0.0f;
  ones.y = ones.x;

  float best[8];
  int   bestk[8];
#pragma unroll
  for (int g = 0; g < 8; ++g) { best[g] = 3.0e38f; bestk[g] = 0; }

  const float* abase = agg + point * DFEAT + hi * 2;

  // Prime buffer 0 with candidate 0.
  v2f buf[2][16];
#pragma unroll
  for (int i = 0; i < 16; ++i)
    buf[0][i] = __builtin_nontemporal_load((const v2f*)(abase + i * 4));

#pragma unroll
  for (int k = 0; k < KCAND; ++k) {
    // Issue next candidate's loads into the other buffer (stay in flight
    // while we compute on the current buffer).
    if (k + 1 < KCAND) {
      const float* anext = abase + (size_t)(k + 1) * kstride;
#pragma unroll
      for (int i = 0; i < 16; ++i)
        buf[(k + 1) & 1][i] =
            __builtin_nontemporal_load((const v2f*)(anext + i * 4));
    }
    if (k + 2 < KCAND)
      __builtin_prefetch(abase + (size_t)(k + 2) * kstride, 0, 1);

    v8f c = {};  // accumulator: D[m][0] = L1 distance of point m to candidate k
#pragma unroll
    for (int i = 0; i < 16; ++i) {
      v2f a = buf[k & 1][i];
      v2f d;
      d.x = __builtin_fabsf(t[i].x - a.x);
      d.y = __builtin_fabsf(t[i].y - a.y);
      // D = A(16x4) x B(4x16) + C ; B ones-column sums 4 features per call.
      c = __builtin_amdgcn_wmma_f32_16x16x4_f32(
          /*neg_a=*/false, d, /*neg_b=*/false, ones,
          /*c_mod=*/(short)0, c, /*reuse_a=*/false, /*reuse_b=*/false);
    }

    // Column 0 lives in lanes 0 (M=0..7) and 16 (M=8..15); other lanes carry
    // junk but the compare is harmless there.
#pragma unroll
    for (int g = 0; g < 8; ++g) {
      float dg = c[g];
      if (dg < best[g]) { best[g] = dg; bestk[g] = k; }
    }
  }

  // Gather agg[kmin, point, 0] and store. Only lanes 0 and 16 are meaningful.
  if (m == 0) {
#pragma unroll
    for (int g = 0; g < 8; ++g) {
      size_t p2 = (size_t)tile * 16 + (size_t)hi * 8 + g;
      out[p2] = agg[(size_t)bestk[g] * kstride + p2 * DFEAT];
    }
  }
}

extern "C" void kernel_launch(void* const* d_in, const int* in_sizes, int n_in,
                              void* d_out, int out_size, void* d_ws, size_t ws_size,
                              hipStream_t stream) {
  const float* tensor = (const float*)d_in[0];      // [B,N,D] f32
  const float* agg    = (const float*)d_in[1];      // [K,B,N,D] f32
  float* out          = (float*)d_out;              // [1,B,N,1] f32

  const int P = in_sizes[0] / DFEAT;                // B*N points
  const int tiles = (P + 15) / 16;                  // 16 points per wave
  const int blocks = (tiles + 7) / 8;               // 8 waves per 256-thread block

  hipLaunchKernelGGL(l1_argmin_gather_kernel, dim3(blocks), dim3(256), 0, stream,
                     tensor, agg, out, P);
}